// FrequenceSelect_27015344292134
// MI455X (gfx1250) — compile-verified
//
#include <hip/hip_runtime.h>
#include <hip/hip_bf16.h>

// FrequenceSelect: rfft -> per-channel complex 257x257 linear (+bias) ->
// softshrink gate -> spectrum * gate -> irfft.
// B=32, S=512, C=512, L=257, Lp=260 (K padded to multiple of 4 with zeros).

#define BATCH 32
#define SEQ 512
#define CH 512
#define LBINS 257
#define LPAD 260
#define LAMBDA 0.01f

typedef __attribute__((ext_vector_type(2))) float v2f;
typedef __attribute__((ext_vector_type(8))) float v8f;

__device__ __forceinline__ unsigned brev9(unsigned x) { return __brev(x) >> 23; }

__device__ __forceinline__ float softshrink(float v) {
    return v > LAMBDA ? v - LAMBDA : (v < -LAMBDA ? v + LAMBDA : 0.0f);
}

// ---------------------------------------------------------------------------
// Kernel 1: forward rFFT. One workgroup = (batch b, 8-channel tile).
// 512-pt radix-2 DIT complex FFT (imag=0 input) in LDS, stride-9 padding.
// Writes Fr/Fi in [C][B][LPAD] layout (zeros in l=257..259 -- this padding is
// what makes the WMMA K-tail clamp in kernel 2 exact).
// ---------------------------------------------------------------------------
__global__ __launch_bounds__(256) void fft_fwd_kernel(const float* __restrict__ x,
                                                      float* __restrict__ Fr,
                                                      float* __restrict__ Fi) {
    __shared__ float sre[SEQ * 9];
    __shared__ float sim[SEQ * 9];
    const int tid = threadIdx.x;
    const int b = blockIdx.x >> 6;        // 64 channel tiles
    const int c0 = (blockIdx.x & 63) * 8;

    // bit-reversed load: LDS position p holds x[b, brev9(p), c0+ch]
    for (int i = tid; i < SEQ * 8; i += 256) {
        const int p = i >> 3, ch = i & 7;
        const int s = (int)brev9((unsigned)p);
        sre[p * 9 + ch] = x[(b * SEQ + s) * CH + c0 + ch];
        sim[p * 9 + ch] = 0.0f;
    }

    for (int m = 1; m <= 9; ++m) {
        __syncthreads();
        const int len = 1 << m, half = len >> 1;
        for (int k = tid; k < 256 * 8; k += 256) {
            const int ch = k & 7;
            const int bf = k >> 3;
            const int j = bf & (half - 1);
            const int grp = bf >> (m - 1);
            const int p1 = grp * len + j;
            const int p2 = p1 + half;
            float sn, cs;
            __sincosf(6.28318530718f * (float)j / (float)len, &sn, &cs);
            const float wr = cs, wi = -sn;     // exp(-i*2*pi*j/len)
            const int a1 = p1 * 9 + ch, a2 = p2 * 9 + ch;
            const float ur = sre[a1], ui = sim[a1];
            const float vr = sre[a2], vi = sim[a2];
            const float tr = vr * wr - vi * wi;
            const float ti = vr * wi + vi * wr;
            sre[a1] = ur + tr; sim[a1] = ui + ti;
            sre[a2] = ur - tr; sim[a2] = ui - ti;
        }
    }
    __syncthreads();

    for (int i = tid; i < LPAD * 8; i += 256) {
        const int l = i >> 3, ch = i & 7;
        const int base = ((c0 + ch) * BATCH + b) * LPAD + l;
        Fr[base] = (l < LBINS) ? sre[l * 9 + ch] : 0.0f;
        Fi[base] = (l < LBINS) ? sim[l * 9 + ch] : 0.0f;
    }
}

// ---------------------------------------------------------------------------
// Kernel 2: per-channel complex GEMM on WMMA f32 16x16x4 + softshrink gate.
// 256-thread blocks = 8 waves; each wave owns one (channel, 16-wide n-tile)
// and covers the FULL M=32 batch (two 16-row M-tiles) so every weight
// fragment feeds 4 WMMAs: 8 loads per 8 WMMAs per K-chunk.
// Branchless inner loop: weight row/col indices are CLAMPED (v_min), not
// guarded -- A-side zero padding guarantees the clamped tail contributes 0.
// Or = Fr@W0 - Fi@W1 + b0 ; Oi = Fi@W0 + Fr@W1 + b1
// gate = softshrink(Or) + i*softshrink(Oi) ; G = F * gate  (complex)
// ---------------------------------------------------------------------------
__global__ __launch_bounds__(256) void wmma_gate_kernel(const float* __restrict__ Fr,
                                                        const float* __restrict__ Fi,
                                                        const float* __restrict__ W0,
                                                        const float* __restrict__ W1,
                                                        const float* __restrict__ b1r,
                                                        const float* __restrict__ b1i,
                                                        float* __restrict__ Gr,
                                                        float* __restrict__ Gi) {
    const int job = blockIdx.x * 8 + threadIdx.y;   // 512*17 = 8704 jobs
    const int c = job / 17;
    const int o0 = (job % 17) * 16;        // n-tile origin (0..256)

    const int lane = threadIdx.x;
    const int half = lane >> 4;            // 0 or 1
    const int lm = lane & 15;

    const int wbase = c * (LBINS * LBINS);
    const int fbase = c * (BATCH * LPAD);

    const int o = o0 + lm;                 // output column for this lane
    const int oc = o < LBINS ? o : (LBINS - 1);   // clamped column (edge tile)

    v8f acc_rr[2] = {{}, {}};              // Fr@W0 per m-tile
    v8f acc_ri[2] = {{}, {}};              // Fr@W1
    v8f acc_ir[2] = {{}, {}};              // Fi@W0
    v8f acc_ii[2] = {{}, {}};              // Fi@W1

    for (int k = 0; k < LPAD; k += 4) {
        const int kb = k + 2 * half;       // this lane's K pair base

        // A fragments (16x4): lane holds (M=lm, K=kb),(M=lm, K=kb+1);
        // two M-tiles (batch rows 0..15 and 16..31)
        const int arow0 = fbase + lm * LPAD + kb;
        const int arow1 = arow0 + 16 * LPAD;
        v2f ar0 = *(const v2f*)(Fr + arow0);
        v2f ai0 = *(const v2f*)(Fi + arow0);
        v2f ar1 = *(const v2f*)(Fr + arow1);
        v2f ai1 = *(const v2f*)(Fi + arow1);

        // B fragments (4x16): clamp rows; A padding zeros the clamped lanes
        const int l0 = kb < LBINS ? kb : (LBINS - 1);
        const int l1 = (kb + 1) < LBINS ? (kb + 1) : (LBINS - 1);
        const int wa = wbase + l0 * LBINS + oc;
        const int wb = wbase + l1 * LBINS + oc;
        v2f bw0, bw1;
        bw0.x = W0[wa]; bw0.y = W0[wb];
        bw1.x = W1[wa]; bw1.y = W1[wb];

        // prefetch weight rows ~4 chunks ahead (global_prefetch_b8)
        {
            const int kp = (k + 16 < LBINS) ? (k + 16) : (LBINS - 1);
            const int wp = wbase + kp * LBINS + oc;
            __builtin_prefetch(&W0[wp], 0, 1);
            __builtin_prefetch(&W1[wp], 0, 1);
        }

        acc_rr[0] = __builtin_amdgcn_wmma_f32_16x16x4_f32(false, ar0, false, bw0,
                                                          (short)0, acc_rr[0], false, false);
        acc_ri[0] = __builtin_amdgcn_wmma_f32_16x16x4_f32(false, ar0, false, bw1,
                                                          (short)0, acc_ri[0], false, false);
        acc_ir[0] = __builtin_amdgcn_wmma_f32_16x16x4_f32(false, ai0, false, bw0,
                                                          (short)0, acc_ir[0], false, false);
        acc_ii[0] = __builtin_amdgcn_wmma_f32_16x16x4_f32(false, ai0, false, bw1,
                                                          (short)0, acc_ii[0], false, false);
        acc_rr[1] = __builtin_amdgcn_wmma_f32_16x16x4_f32(false, ar1, false, bw0,
                                                          (short)0, acc_rr[1], false, false);
        acc_ri[1] = __builtin_amdgcn_wmma_f32_16x16x4_f32(false, ar1, false, bw1,
                                                          (short)0, acc_ri[1], false, false);
        acc_ir[1] = __builtin_amdgcn_wmma_f32_16x16x4_f32(false, ai1, false, bw0,
                                                          (short)0, acc_ir[1], false, false);
        acc_ii[1] = __builtin_amdgcn_wmma_f32_16x16x4_f32(false, ai1, false, bw1,
                                                          (short)0, acc_ii[1], false, false);
    }

    // epilogue: bias, softshrink gate, complex multiply with spectrum, store
    if (o >= LBINS) return;               // edge-tile lanes: nothing to store
    const float bb0 = b1r[c * LBINS + o];
    const float bb1 = b1i[c * LBINS + o];

    for (int mt = 0; mt < 2; ++mt) {
        for (int j = 0; j < 8; ++j) {
            const int bidx = mt * 16 + j + half * 8;     // batch row (C/D layout)
            const float Orr = acc_rr[mt][j] - acc_ii[mt][j] + bb0;
            const float Oii = acc_ir[mt][j] + acc_ri[mt][j] + bb1;
            const float gr = softshrink(Orr);
            const float gi = softshrink(Oii);
            const int fidx = fbase + bidx * LPAD + o;
            const float fr = Fr[fidx];
            const float fi = Fi[fidx];
            Gr[fidx] = fr * gr - fi * gi;
            Gi[fidx] = fr * gi + fi * gr;
        }
    }
}

// ---------------------------------------------------------------------------
// Kernel 3: inverse FFT. Hermitian-extend G to 512 bins, inverse DIT in LDS,
// real part / 512 -> out[b, s, c].
// ---------------------------------------------------------------------------
__global__ __launch_bounds__(256) void fft_inv_kernel(const float* __restrict__ Gr,
                                                      const float* __restrict__ Gi,
                                                      float* __restrict__ out) {
    __shared__ float sre[SEQ * 9];
    __shared__ float sim[SEQ * 9];
    const int tid = threadIdx.x;
    const int b = blockIdx.x >> 6;
    const int c0 = (blockIdx.x & 63) * 8;

    for (int i = tid; i < SEQ * 8; i += 256) {
        const int p = i >> 3, ch = i & 7;
        const int s = (int)brev9((unsigned)p);
        const int l = (s <= 256) ? s : SEQ - s;
        const float sgn = (s <= 256) ? 1.0f : -1.0f;
        const int base = ((c0 + ch) * BATCH + b) * LPAD + l;
        sre[p * 9 + ch] = Gr[base];
        sim[p * 9 + ch] = sgn * Gi[base];
    }

    for (int m = 1; m <= 9; ++m) {
        __syncthreads();
        const int len = 1 << m, half = len >> 1;
        for (int k = tid; k < 256 * 8; k += 256) {
            const int ch = k & 7;
            const int bf = k >> 3;
            const int j = bf & (half - 1);
            const int grp = bf >> (m - 1);
            const int p1 = grp * len + j;
            const int p2 = p1 + half;
            float sn, cs;
            __sincosf(6.28318530718f * (float)j / (float)len, &sn, &cs);
            const float wr = cs, wi = sn;      // exp(+i*2*pi*j/len)
            const int a1 = p1 * 9 + ch, a2 = p2 * 9 + ch;
            const float ur = sre[a1], ui = sim[a1];
            const float vr = sre[a2], vi = sim[a2];
            const float tr = vr * wr - vi * wi;
            const float ti = vr * wi + vi * wr;
            sre[a1] = ur + tr; sim[a1] = ui + ti;
            sre[a2] = ur - tr; sim[a2] = ui - ti;
        }
    }
    __syncthreads();

    for (int i = tid; i < SEQ * 8; i += 256) {
        const int s = i >> 3, ch = i & 7;
        out[((long)b * SEQ + s) * CH + c0 + ch] = sre[s * 9 + ch] * (1.0f / 512.0f);
    }
}

extern "C" void kernel_launch(void* const* d_in, const int* in_sizes, int n_in,
                              void* d_out, int out_size, void* d_ws, size_t ws_size,
                              hipStream_t stream) {
    const float* x  = (const float*)d_in[0];   // [32,512,512]
    const float* w1 = (const float*)d_in[1];   // [2,512,257,257]
    const float* b1 = (const float*)d_in[2];   // [2,512,257]
    float* out = (float*)d_out;                // [32,512,512]

    // split real/imag halves on the host so the kernel gets clean SGPR bases
    const float* W0  = w1;
    const float* W1  = w1 + (long)CH * LBINS * LBINS;
    const float* b1r = b1;
    const float* b1i = b1 + (long)CH * LBINS;

    // workspace: Fr, Fi, Gr, Gi each [C][B][LPAD] floats (~17 MB each)
    const long NF = (long)CH * BATCH * LPAD;
    float* ws = (float*)d_ws;
    float* Fr = ws;
    float* Fi = ws + NF;
    float* Gr = ws + 2 * NF;
    float* Gi = ws + 3 * NF;
    (void)ws_size; (void)in_sizes; (void)n_in; (void)out_size;

    // 1) forward FFT: 32 batches x 64 channel-tiles
    fft_fwd_kernel<<<dim3(BATCH * 64), dim3(256), 0, stream>>>(x, Fr, Fi);

    // 2) WMMA complex linear + gate: 8704 wave-jobs (c x n-tile), 8 waves/block
    wmma_gate_kernel<<<dim3((CH * 17) / 8), dim3(32, 8), 0, stream>>>(
        Fr, Fi, W0, W1, b1r, b1i, Gr, Gi);

    // 3) inverse FFT
    fft_inv_kernel<<<dim3(BATCH * 64), dim3(256), 0, stream>>>(Gr, Gi, out);
}